// GAT_87428354278169
// MI455X (gfx1250) — compile-verified
//
#include <hip/hip_runtime.h>

typedef float v2f __attribute__((ext_vector_type(2)));
typedef float v8f __attribute__((ext_vector_type(8)));

#define GAT_SLOPE 0.2f

__device__ __forceinline__ float gat_leaky(float x) { return x > 0.0f ? x : GAT_SLOPE * x; }

// Ordered-key mapping so unsigned atomicMax == float max (handles negatives).
__device__ __forceinline__ unsigned f2key(float f) {
    unsigned b = __float_as_uint(f);
    return (b & 0x80000000u) ? ~b : (b | 0x80000000u);
}
__device__ __forceinline__ float key2f(unsigned k) {
    unsigned b = (k & 0x80000000u) ? (k & 0x7fffffffu) : ~k;
    return __uint_as_float(b);
}

// ---------------------------------------------------------------------------
// C[N,128] = A[N,K] @ W[K,128], f32, full precision via V_WMMA_F32_16X16X4_F32.
// Block = 256 threads = 8 waves; wave w computes 16x16 tile at column w*16.
// Grid.x = ceil(N/16). K is a multiple of 4 (64 or 128 here).
// A-matrix 16x4 f32 layout:  lane m=lane&15, half=lane>>4: v[0]=A[m][k+2*half],
// v[1]=A[m][k+2*half+1].  B 4x16 mirrors with n=lane&15.  C/D: VGPR v holds
// row (half*8+v), col lane&15.
// ---------------------------------------------------------------------------
__global__ void gat_gemm128_wmma(const float* __restrict__ A,
                                 const float* __restrict__ W,
                                 float* __restrict__ C, int N, int K) {
    const int lane = threadIdx.x & 31;
    const int wave = threadIdx.x >> 5;      // 0..7 -> column tile
    const int m    = lane & 15;
    const int half = lane >> 4;             // 0 or 1
    int row  = blockIdx.x * 16 + m;
    int rowc = row < N ? row : (N - 1);     // clamp loads (WMMA needs full EXEC)
    const int col = wave * 16 + m;

    const float* Arow = A + (size_t)rowc * K;
    v8f acc = {0.f, 0.f, 0.f, 0.f, 0.f, 0.f, 0.f, 0.f};

    for (int k = 0; k < K; k += 4) {
        const int ka = k + half * 2;
        v2f a, b;
        a[0] = Arow[ka];
        a[1] = Arow[ka + 1];
        b[0] = W[(size_t)ka * 128 + col];
        b[1] = W[(size_t)(ka + 1) * 128 + col];
        acc = __builtin_amdgcn_wmma_f32_16x16x4_f32(false, a, false, b,
                                                    (short)0, acc, false, false);
    }

    const int rbase = blockIdx.x * 16 + half * 8;
    float* Cp = C + (size_t)rbase * 128 + wave * 16 + m;
#pragma unroll
    for (int v = 0; v < 8; ++v) {
        if (rbase + v < N) Cp[(size_t)v * 128] = acc[v];
    }
}

// out[n*4+h] = dot(Hm[n, h*32 : h*32+32], avec[h*32 : h*32+32])
__global__ void gat_attn_dot(const float* __restrict__ Hm,
                             const float* __restrict__ avec,
                             float* __restrict__ out, int N) {
    int idx = blockIdx.x * blockDim.x + threadIdx.x;
    if (idx >= N * 4) return;
    int n = idx >> 2, h = idx & 3;
    const float* hp = Hm + (size_t)n * 128 + h * 32;
    const float* ap = avec + h * 32;
    float s = 0.f;
#pragma unroll
    for (int j = 0; j < 32; j += 4) {
        float4 hv = *(const float4*)(hp + j);
        float4 av = *(const float4*)(ap + j);
        s += hv.x * av.x + hv.y * av.y + hv.z * av.z + hv.w * av.w;
    }
    out[idx] = s;
}

__global__ void gat_fill_u32(unsigned* __restrict__ p, unsigned val, int n) {
    int i = blockIdx.x * blockDim.x + threadIdx.x;
    if (i < n) p[i] = val;
}

// out[n,c] = b[c]  (initialize accumulator with bias)
__global__ void gat_init_bias(float* __restrict__ out, const float* __restrict__ b, int N) {
    int i = blockIdx.x * blockDim.x + threadIdx.x;
    if (i < N * 128) out[i] = b[i & 127];
}

// Pass 1: segment max of leaky(el[src]+er[dst]) into mkey (ordered-uint).
__global__ void gat_edge_max(const int* __restrict__ src, const int* __restrict__ dst,
                             const float* __restrict__ el, const float* __restrict__ er,
                             unsigned* __restrict__ mkey, int E) {
    int e = blockIdx.x * blockDim.x + threadIdx.x;
    if (e >= E) return;
    int s = src[e], d = dst[e];
    float4 l = *(const float4*)(el + (size_t)s * 4);
    float4 r = *(const float4*)(er + (size_t)d * 4);
    atomicMax(&mkey[(size_t)d * 4 + 0], f2key(gat_leaky(l.x + r.x)));
    atomicMax(&mkey[(size_t)d * 4 + 1], f2key(gat_leaky(l.y + r.y)));
    atomicMax(&mkey[(size_t)d * 4 + 2], f2key(gat_leaky(l.z + r.z)));
    atomicMax(&mkey[(size_t)d * 4 + 3], f2key(gat_leaky(l.w + r.w)));
}

// Pass 2: denom[dst,h] += exp(e - m[dst,h])
__global__ void gat_edge_expsum(const int* __restrict__ src, const int* __restrict__ dst,
                                const float* __restrict__ el, const float* __restrict__ er,
                                const unsigned* __restrict__ mkey,
                                float* __restrict__ denom, int E) {
    int e = blockIdx.x * blockDim.x + threadIdx.x;
    if (e >= E) return;
    int s = src[e], d = dst[e];
    float4 l = *(const float4*)(el + (size_t)s * 4);
    float4 r = *(const float4*)(er + (size_t)d * 4);
#pragma unroll
    for (int h = 0; h < 4; ++h) {
        float ev = gat_leaky(((const float*)&l)[h] + ((const float*)&r)[h]);
        float ex = expf(ev - key2f(mkey[(size_t)d * 4 + h]));
        atomicAdd(&denom[(size_t)d * 4 + h], ex);
    }
}

// Pass 3: out[dst] += alpha * hs[src].  One wave per edge; lane handles 4
// contiguous columns (float4 gather -> fully coalesced 512B per wave).
__global__ void gat_edge_agg(const int* __restrict__ src, const int* __restrict__ dst,
                             const float* __restrict__ el, const float* __restrict__ er,
                             const unsigned* __restrict__ mkey,
                             const float* __restrict__ denom,
                             const float* __restrict__ hs,
                             float* __restrict__ out, int E) {
    int lane = threadIdx.x & 31;
    int e = blockIdx.x * (blockDim.x >> 5) + (threadIdx.x >> 5);
    if (e >= E) return;
    int s = src[e], d = dst[e];
    int h = lane >> 3;                       // 4 cols/lane -> head = lane/8
    float ev = gat_leaky(el[(size_t)s * 4 + h] + er[(size_t)d * 4 + h]);
    float alpha = expf(ev - key2f(mkey[(size_t)d * 4 + h])) / denom[(size_t)d * 4 + h];
    float4 hv = *(const float4*)(hs + (size_t)s * 128 + lane * 4);
    float* op = out + (size_t)d * 128 + lane * 4;
    atomicAdd(op + 0, alpha * hv.x);
    atomicAdd(op + 1, alpha * hv.y);
    atomicAdd(op + 2, alpha * hv.z);
    atomicAdd(op + 3, alpha * hv.w);
}

// ---------------------------------------------------------------------------
static void run_conv(const float* src_feat, int Nsrc, const float* dst_feat, int Ndst,
                     int Kin, const float* Wm, const float* al, const float* ar,
                     const float* bias, const int* e_src, const int* e_dst, int E,
                     float* hs, float* hd, float* el, float* er,
                     unsigned* mkey, float* denom, float* out, hipStream_t stream) {
    gat_gemm128_wmma<<<dim3((Nsrc + 15) / 16), dim3(256), 0, stream>>>(src_feat, Wm, hs, Nsrc, Kin);
    gat_gemm128_wmma<<<dim3((Ndst + 15) / 16), dim3(256), 0, stream>>>(dst_feat, Wm, hd, Ndst, Kin);
    gat_attn_dot<<<dim3((Nsrc * 4 + 255) / 256), dim3(256), 0, stream>>>(hs, al, el, Nsrc);
    gat_attn_dot<<<dim3((Ndst * 4 + 255) / 256), dim3(256), 0, stream>>>(hd, ar, er, Ndst);
    gat_fill_u32<<<dim3((Ndst * 4 + 255) / 256), dim3(256), 0, stream>>>(mkey, 0u, Ndst * 4);
    gat_fill_u32<<<dim3((Ndst * 4 + 255) / 256), dim3(256), 0, stream>>>((unsigned*)denom, 0u, Ndst * 4);
    gat_init_bias<<<dim3((Ndst * 128 + 255) / 256), dim3(256), 0, stream>>>(out, bias, Ndst);
    gat_edge_max<<<dim3((E + 255) / 256), dim3(256), 0, stream>>>(e_src, e_dst, el, er, mkey, E);
    gat_edge_expsum<<<dim3((E + 255) / 256), dim3(256), 0, stream>>>(e_src, e_dst, el, er, mkey, denom, E);
    gat_edge_agg<<<dim3((E + 7) / 8), dim3(256), 0, stream>>>(e_src, e_dst, el, er, mkey, denom, hs, out, E);
}

extern "C" void kernel_launch(void* const* d_in, const int* in_sizes, int n_in,
                              void* d_out, int out_size, void* d_ws, size_t ws_size,
                              hipStream_t stream) {
    const float* user_feat = (const float*)d_in[0];
    const float* sub_feat  = (const float*)d_in[1];
    const float* W1_i  = (const float*)d_in[2];
    const float* al1_i = (const float*)d_in[3];
    const float* ar1_i = (const float*)d_in[4];
    const float* b1_i  = (const float*)d_in[5];
    const float* W1_r  = (const float*)d_in[6];
    const float* al1_r = (const float*)d_in[7];
    const float* ar1_r = (const float*)d_in[8];
    const float* b1_r  = (const float*)d_in[9];
    const float* W2_i  = (const float*)d_in[10];
    const float* al2_i = (const float*)d_in[11];
    const float* ar2_i = (const float*)d_in[12];
    const float* b2_i  = (const float*)d_in[13];
    const float* W2_r  = (const float*)d_in[14];
    const float* al2_r = (const float*)d_in[15];
    const float* ar2_r = (const float*)d_in[16];
    const float* b2_r  = (const float*)d_in[17];
    const int* src_user = (const int*)d_in[18];
    const int* dst_sub  = (const int*)d_in[19];

    const int FH = 128;                         // H * HID = H * OUT
    const int IN = in_sizes[2] / FH;            // 64
    const int NU = in_sizes[0] / IN;            // 200000
    const int NS = in_sizes[1] / IN;            // 20000
    const int E  = in_sizes[18];                // 1000000

    // Workspace layout (f32 units)
    float* ws = (float*)d_ws;
    size_t p = 0;
    float* tmpU   = ws + p; p += (size_t)NU * FH;   // projection temp (user-sized)
    float* tmpS   = ws + p; p += (size_t)NS * FH;   // projection temp (sub-sized)
    float* user_h = ws + p; p += (size_t)NU * FH;   // layer-1 user output
    float* sub_h  = ws + p; p += (size_t)NS * FH;   // layer-1 sub output
    float* el     = ws + p; p += (size_t)NU * 4;
    float* er     = ws + p; p += (size_t)NU * 4;
    unsigned* mkey = (unsigned*)(ws + p); p += (size_t)NU * 4;
    float* denom  = ws + p; p += (size_t)NU * 4;

    float* user_o = (float*)d_out;                  // [NU,128]
    float* sub_o  = user_o + (size_t)NU * FH;       // [NS,128]

    // Layer 1, relation user->sub (interacts): out = sub_h
    run_conv(user_feat, NU, sub_feat, NS, IN, W1_i, al1_i, ar1_i, b1_i,
             src_user, dst_sub, E, tmpU, tmpS, el, er, mkey, denom, sub_h, stream);
    // Layer 1, reverse relation sub->user: out = user_h
    run_conv(sub_feat, NS, user_feat, NU, IN, W1_r, al1_r, ar1_r, b1_r,
             dst_sub, src_user, E, tmpS, tmpU, el, er, mkey, denom, user_h, stream);
    // Layer 2, user_h -> sub_h edges: out = sub_o
    run_conv(user_h, NU, sub_h, NS, FH, W2_i, al2_i, ar2_i, b2_i,
             src_user, dst_sub, E, tmpU, tmpS, el, er, mkey, denom, sub_o, stream);
    // Layer 2, reverse: out = user_o
    run_conv(sub_h, NS, user_h, NU, FH, W2_r, al2_r, ar2_r, b2_r,
             dst_sub, src_user, E, tmpS, tmpU, el, er, mkey, denom, user_o, stream);
}